// AttentionModel_28278064676954
// MI455X (gfx1250) — compile-verified
//
#include <hip/hip_runtime.h>
#include <hip/hip_bf16.h>

// Problem dimensions (compile-time constants from the reference)
constexpr int N_SEQ = 1024;
constexpr int L     = 64;
constexpr int DH    = 256;
constexpr int DX    = 128;
constexpr int M_Q   = 8192;
constexpr int G     = 4096;
constexpr int N_TYP = 64;
constexpr int N_MEM = 262144;

typedef __attribute__((ext_vector_type(2))) float v2f;
typedef __attribute__((ext_vector_type(8))) float v8f;

// ---------------------------------------------------------------------------
// Stage 1a: zero the segment-sum accumulator z (G x DX)
// ---------------------------------------------------------------------------
__global__ void zero_f32_kernel(float* __restrict__ p, int n) {
    int i = blockIdx.x * blockDim.x + threadIdx.x;
    if (i < n) p[i] = 0.0f;
}

// ---------------------------------------------------------------------------
// Stage 1b: z[grp[i], :] += tok_emb[mem[i], :]   (scatter-add, atomics)
// one thread per (memory entry, feature) -> coalesced reads and atomics
// ---------------------------------------------------------------------------
__global__ void segment_sum_kernel(const int* __restrict__ mem,
                                   const int* __restrict__ grp,
                                   const float* __restrict__ tok_emb,
                                   float* __restrict__ z) {
    long gid = (long)blockIdx.x * blockDim.x + threadIdx.x;
    if (gid >= (long)N_MEM * DX) return;
    int i = (int)(gid >> 7);      // / DX
    int v = (int)(gid & (DX - 1));
    int tok = mem[i];
    int g   = grp[i];
    atomicAdd(&z[g * DX + v], tok_emb[tok * DX + v]);
}

// ---------------------------------------------------------------------------
// Stage 2 helper: generic fp32 WMMA GEMM, D = A(MxK) * B(KxN) + bias
// One wave computes one 16x16 tile of D using V_WMMA_F32_16X16X4_F32.
// A-operand layout (16x4 f32): lanes 0-15 -> K={k,k+1}, lanes 16-31 -> K={k+2,k+3}
// B-operand layout (4x16 f32): same K split, N = lane & 15
// C/D (16x16 f32, 8 VGPRs): VGPR r -> row r + 8*(lane>=16), col = lane & 15
// ---------------------------------------------------------------------------
__global__ void gemm_wmma_f32_kernel(const float* __restrict__ A,
                                     const float* __restrict__ B,
                                     const float* __restrict__ bias,
                                     float* __restrict__ D,
                                     int Mrows, int N, int K) {
    int wavesPerBlock = blockDim.x >> 5;
    int wave  = blockIdx.x * wavesPerBlock + (threadIdx.x >> 5);
    int ntile = N >> 4;
    int tileM = (wave / ntile) << 4;
    int tileN = (wave % ntile) << 4;
    if (tileM >= Mrows) return;

    int lane = threadIdx.x & 31;
    int half = lane >> 4;      // 0: K pair {0,1}; 1: K pair {2,3}
    int mr   = lane & 15;      // A row / B-D column within tile

    v8f c = {};
    const long arow = (long)(tileM + mr) * K;
    for (int k = 0; k < K; k += 4) {
        int ka = k + 2 * half;
        v2f a, b;
        a.x = A[arow + ka];
        a.y = A[arow + ka + 1];
        b.x = B[(long)ka * N + tileN + mr];
        b.y = B[(long)(ka + 1) * N + tileN + mr];
        c = __builtin_amdgcn_wmma_f32_16x16x4_f32(
                /*neg_a=*/false, a, /*neg_b=*/false, b,
                /*c_mod=*/(short)0, c, /*reuse_a=*/false, /*reuse_b=*/false);
    }
    float bv = bias[tileN + mr];
#pragma unroll
    for (int r = 0; r < 8; ++r) {
        int row = tileM + r + 8 * half;
        D[(long)row * N + tileN + mr] = c[r] + bv;
    }
}

// ---------------------------------------------------------------------------
// Stage 3a: build q (M x 2*DH) = concat(h_grp[idx,src], h_grp[idx,dst])
// ---------------------------------------------------------------------------
__global__ void build_q_kernel(const float* __restrict__ h_grp,
                               const int* __restrict__ idx,
                               const int* __restrict__ src,
                               const int* __restrict__ dst,
                               float* __restrict__ q) {
    int m = blockIdx.x;
    int t = threadIdx.x;  // 256 threads == DH
    int seq = idx[m];
    long srow = ((long)seq * L + src[m]) * DH;
    long drow = ((long)seq * L + dst[m]) * DH;
    q[(long)m * (2 * DH) + t]      = h_grp[srow + t];
    q[(long)m * (2 * DH) + DH + t] = h_grp[drow + t];
}

// ---------------------------------------------------------------------------
// Stage 4: per-query attention: scores -> masked softmax -> ctx (M x DX)
// Block = 128 threads per query. kp (64 MB) lives in L2 (192 MB) across reuse.
// ---------------------------------------------------------------------------
__global__ void attention_kernel(const float* __restrict__ kp,
                                 const float* __restrict__ qp,
                                 const float* __restrict__ z,
                                 const int* __restrict__ idx,
                                 const int* __restrict__ pos2grp,
                                 const int* __restrict__ msk,
                                 float* __restrict__ ctx) {
    __shared__ float qs[DH];
    __shared__ float sc[L];
    __shared__ float sinv;

    int m = blockIdx.x;
    int t = threadIdx.x;      // 128 threads
    int seq = idx[m];

    qs[t]       = qp[(long)m * DH + t];
    qs[t + 128] = qp[(long)m * DH + 128 + t];
    __syncthreads();

    if (t < L) {
        const float* krow = kp + ((long)seq * L + t) * DH;
        float acc = 0.0f;
#pragma unroll 8
        for (int d = 0; d < DH; ++d) acc += krow[d] * qs[d];
        acc *= 0.0625f;  // 1/sqrt(DH)
        sc[t] = msk[seq * L + t] ? acc : -__builtin_inff();
    }
    __syncthreads();

    if (t == 0) {
        float mx = sc[0];
        for (int l = 1; l < L; ++l) mx = fmaxf(mx, sc[l]);
        float sum = 0.0f;
        for (int l = 0; l < L; ++l) {
            float e = expf(sc[l] - mx);   // exp(-inf) == 0 for masked slots
            sc[l] = e;
            sum += e;
        }
        sinv = 1.0f / sum;
    }
    __syncthreads();

    float inv = sinv;
    float acc = 0.0f;
    for (int l = 0; l < L; ++l) {
        int g = pos2grp[seq * L + l];
        acc += sc[l] * z[(long)g * DX + t];
    }
    ctx[(long)m * DX + t] = acc * inv;
}

// ---------------------------------------------------------------------------
// Stage 5: logit = concat(q, ctx) @ Wrel + brel  (K = 2*DH + DX = 640, N = 64)
// Same WMMA tiling; A-operand fetched virtually from q / ctx (no concat buf).
// ---------------------------------------------------------------------------
__global__ void logit_wmma_kernel(const float* __restrict__ q,
                                  const float* __restrict__ ctx,
                                  const float* __restrict__ Wrel,
                                  const float* __restrict__ brel,
                                  float* __restrict__ out) {
    constexpr int K = 2 * DH + DX;   // 640
    constexpr int N = N_TYP;         // 64
    int wavesPerBlock = blockDim.x >> 5;
    int wave  = blockIdx.x * wavesPerBlock + (threadIdx.x >> 5);
    int ntile = N >> 4;              // 4
    int tileM = (wave / ntile) << 4;
    int tileN = (wave % ntile) << 4;
    if (tileM >= M_Q) return;

    int lane = threadIdx.x & 31;
    int half = lane >> 4;
    int mr   = lane & 15;
    int row  = tileM + mr;

    v8f c = {};
    for (int k = 0; k < K; k += 4) {
        int ka = k + 2 * half;       // ka, ka+1 never straddle the 512 boundary
        v2f a, b;
        if (ka < 2 * DH) {
            a.x = q[(long)row * (2 * DH) + ka];
            a.y = q[(long)row * (2 * DH) + ka + 1];
        } else {
            a.x = ctx[(long)row * DX + (ka - 2 * DH)];
            a.y = ctx[(long)row * DX + (ka - 2 * DH) + 1];
        }
        b.x = Wrel[(long)ka * N + tileN + mr];
        b.y = Wrel[(long)(ka + 1) * N + tileN + mr];
        c = __builtin_amdgcn_wmma_f32_16x16x4_f32(
                false, a, false, b, (short)0, c, false, false);
    }
    float bv = brel[tileN + mr];
#pragma unroll
    for (int r = 0; r < 8; ++r) {
        out[(long)(tileM + r + 8 * half) * N + tileN + mr] = c[r] + bv;
    }
}

// ---------------------------------------------------------------------------
extern "C" void kernel_launch(void* const* d_in, const int* in_sizes, int n_in,
                              void* d_out, int out_size, void* d_ws, size_t ws_size,
                              hipStream_t stream) {
    const int*   mem     = (const int*)d_in[0];
    const int*   grp     = (const int*)d_in[1];
    const int*   pos2grp = (const int*)d_in[2];
    const float* h_grp   = (const float*)d_in[3];
    const int*   msk     = (const int*)d_in[4];
    const int*   idx     = (const int*)d_in[5];
    const int*   src     = (const int*)d_in[6];
    const int*   dst     = (const int*)d_in[7];
    /* d_in[8] = typ: unused by the reference computation */
    const float* tok_emb = (const float*)d_in[9];
    const float* Wq      = (const float*)d_in[10];
    const float* bq      = (const float*)d_in[11];
    const float* Wk      = (const float*)d_in[12];
    const float* bk      = (const float*)d_in[13];
    const float* Wrel    = (const float*)d_in[14];
    const float* brel    = (const float*)d_in[15];
    float* out = (float*)d_out;

    // Workspace layout (fp32 elements)
    float* ws  = (float*)d_ws;
    float* z   = ws;                              // G*DX        =   524288
    float* kp  = z   + (long)G * DX;              // N_SEQ*L*DH  = 16777216
    float* q   = kp  + (long)N_SEQ * L * DH;      // M*2*DH      =  4194304
    float* qp  = q   + (long)M_Q * 2 * DH;        // M*DH        =  2097152
    float* ctx = qp  + (long)M_Q * DH;            // M*DX        =  1048576

    // 1) z = segment_sum(tok_emb[mem], grp)
    {
        int n = G * DX;
        zero_f32_kernel<<<(n + 255) / 256, 256, 0, stream>>>(z, n);
        long total = (long)N_MEM * DX;
        segment_sum_kernel<<<(int)((total + 255) / 256), 256, 0, stream>>>(
            mem, grp, tok_emb, z);
    }

    // 2) kp = h_grp @ Wk + bk   -- (65536 x 256) x (256 x 256)
    {
        int Mr = N_SEQ * L, N = DH, K = DH;
        int tiles = (Mr >> 4) * (N >> 4);         // 65536 waves
        gemm_wmma_f32_kernel<<<tiles / 8, 256, 0, stream>>>(h_grp, Wk, bk, kp, Mr, N, K);
    }

    // 3) q gather + qp = q @ Wq + bq  -- (8192 x 512) x (512 x 256)
    build_q_kernel<<<M_Q, 256, 0, stream>>>(h_grp, idx, src, dst, q);
    {
        int Mr = M_Q, N = DH, K = 2 * DH;
        int tiles = (Mr >> 4) * (N >> 4);         // 8192 waves
        gemm_wmma_f32_kernel<<<tiles / 8, 256, 0, stream>>>(q, Wq, bq, qp, Mr, N, K);
    }

    // 4) masked softmax attention -> ctx
    attention_kernel<<<M_Q, 128, 0, stream>>>(kp, qp, z, idx, pos2grp, msk, ctx);

    // 5) logit = [q | ctx] @ Wrel + brel -> d_out  (8192 x 64)
    {
        int tiles = (M_Q >> 4) * (N_TYP >> 4);    // 2048 waves
        logit_wmma_kernel<<<tiles / 8, 256, 0, stream>>>(q, ctx, Wrel, brel, out);
    }
}